// TAGConv_50783693308333
// MI455X (gfx1250) — compile-verified
//
#include <hip/hip_runtime.h>

typedef __attribute__((ext_vector_type(2))) float v2f;
typedef __attribute__((ext_vector_type(8))) float v8f;

// ---------------- degree / normalization ----------------

__global__ void tag_init_deg(float* __restrict__ deg, int n) {
  int i = blockIdx.x * blockDim.x + threadIdx.x;
  if (i < n) deg[i] = 1.0f;  // self loop
}

__global__ void tag_deg_scatter(const int* __restrict__ dst, float* __restrict__ deg, int e) {
  int i = blockIdx.x * blockDim.x + threadIdx.x;
  if (i < e) atomicAdd(&deg[dst[i]], 1.0f);
}

__global__ void tag_dinv(const float* __restrict__ deg, float* __restrict__ dinv, int n) {
  int i = blockIdx.x * blockDim.x + threadIdx.x;
  if (i < n) dinv[i] = rsqrtf(deg[i]);  // deg >= 1 always
}

// ---------------- SpMM hop ----------------
// Self-loop term doubles as zero-init: h_new[i,:] = dinv[i]^2 * h_old[i,:]
__global__ void tag_selfloop(const float* __restrict__ h_old, const float* __restrict__ dinv,
                             float* __restrict__ h_new, int n) {
  int t = blockIdx.x * blockDim.x + threadIdx.x;
  int node = t >> 4;
  int f = (t & 15) << 2;
  if (node < n) {
    float w = dinv[node];
    w *= w;
    const float4 v = *(const float4*)(h_old + (size_t)node * 64 + f);
    float4 o;
    o.x = w * v.x; o.y = w * v.y; o.z = w * v.z; o.w = w * v.w;
    *(float4*)(h_new + (size_t)node * 64 + f) = o;
  }
}

// 16 lanes per edge, 4 features per lane: float4 gather (L2-resident) + f32 atomics.
__global__ void tag_spmm(const int* __restrict__ src, const int* __restrict__ dst,
                         const float* __restrict__ dinv, const float* __restrict__ h_old,
                         float* __restrict__ h_new, int e) {
  int t = blockIdx.x * blockDim.x + threadIdx.x;
  int edge = t >> 4;
  int f = (t & 15) << 2;
  if (edge < e) {
    int s = src[edge];
    int d = dst[edge];
    float w = dinv[s] * dinv[d];
    const float4 v = *(const float4*)(h_old + (size_t)s * 64 + f);
    float* o = h_new + (size_t)d * 64 + f;
    atomicAdd(o + 0, w * v.x);
    atomicAdd(o + 1, w * v.y);
    atomicAdd(o + 2, w * v.z);
    atomicAdd(o + 3, w * v.w);
  }
}

// ---------------- dense tail: [x|h1|h2] @ W + b via V_WMMA_F32_16X16X4_F32 ----------------
// One wave32 computes a 16x64 output tile: 4 accumulators over N, 48 k-steps of 4.
__global__ __launch_bounds__(256)
void tag_gemm(const float* __restrict__ x, const float* __restrict__ h1,
              const float* __restrict__ h2, const float* __restrict__ W,
              const float* __restrict__ bias, float* __restrict__ out, int n) {
  const int wave = (int)((blockIdx.x * (unsigned)blockDim.x + threadIdx.x) >> 5);
  const int lane = threadIdx.x & 31;
  const int m0 = wave << 4;
  if (m0 >= n) return;  // wave-uniform: EXEC stays all-1s for WMMA

  const int hi = lane >> 4;      // 0 = lanes 0-15, 1 = lanes 16-31
  const int khalf = hi << 1;     // A/B: vgpr pair covers K = {0,1} (lo lanes) / {2,3} (hi lanes)
  const int col = lane & 15;

  int row = m0 + (lane & 15);
  if (row >= n) row = n - 1;     // clamp without EXEC change (n % 16 == 0 here anyway)

  v8f acc0 = {}; v8f acc1 = {}; v8f acc2 = {}; v8f acc3 = {};

  const float* srcs[3] = {x, h1, h2};
#pragma unroll
  for (int blk = 0; blk < 3; ++blk) {
    const float* __restrict__ A  = srcs[blk] + (size_t)row * 64;
    const float* __restrict__ Wb = W + (size_t)blk * 64 * 64 + col;
#pragma unroll 4
    for (int k = 0; k < 64; k += 4) {
      // A tile 16x4: lane (hi,m) holds A[m][k+khalf], A[m][k+khalf+1]
      v2f a;
      a.x = A[k + khalf];
      a.y = A[k + khalf + 1];
      // B tiles 4x16 for n0 = 0,16,32,48: lane (hi,ncol) holds W[k+khalf][n], W[k+khalf+1][n]
      const float* w0 = Wb + (size_t)(k + khalf) * 64;
      v2f b0; b0.x = w0[0];  b0.y = w0[64];
      v2f b1; b1.x = w0[16]; b1.y = w0[80];
      v2f b2; b2.x = w0[32]; b2.y = w0[96];
      v2f b3; b3.x = w0[48]; b3.y = w0[112];
      acc0 = __builtin_amdgcn_wmma_f32_16x16x4_f32(false, a, false, b0, (short)0, acc0, false, false);
      acc1 = __builtin_amdgcn_wmma_f32_16x16x4_f32(false, a, false, b1, (short)0, acc1, false, false);
      acc2 = __builtin_amdgcn_wmma_f32_16x16x4_f32(false, a, false, b2, (short)0, acc2, false, false);
      acc3 = __builtin_amdgcn_wmma_f32_16x16x4_f32(false, a, false, b3, (short)0, acc3, false, false);
    }
  }

  const float bi0 = bias[col], bi1 = bias[16 + col], bi2 = bias[32 + col], bi3 = bias[48 + col];
#pragma unroll
  for (int r = 0; r < 8; ++r) {
    int m = m0 + r + (hi << 3);  // C/D layout: vgpr r -> M = r + 8*hi
    if (m < n) {
      float* o = out + (size_t)m * 64 + col;
      o[0]  = acc0[r] + bi0;
      o[16] = acc1[r] + bi1;
      o[32] = acc2[r] + bi2;
      o[48] = acc3[r] + bi3;
    }
  }
}

// ---------------- launch ----------------

extern "C" void kernel_launch(void* const* d_in, const int* in_sizes, int n_in,
                              void* d_out, int out_size, void* d_ws, size_t ws_size,
                              hipStream_t stream) {
  const float* x    = (const float*)d_in[0];
  const int*   eidx = (const int*)d_in[1];
  const float* W    = (const float*)d_in[2];
  const float* b    = (const float*)d_in[3];
  float*       out  = (float*)d_out;

  const int n = in_sizes[0] / 64;   // 100000
  const int e = in_sizes[1] / 2;    // 1600000
  const int* src = eidx;            // edge_index[0]
  const int* dst = eidx + e;        // edge_index[1]

  // workspace layout: deg[n] | dinv[n] | h1[n*64] | h2[n*64]  (~52 MB)
  float* ws   = (float*)d_ws;
  float* deg  = ws;
  float* dinv = ws + (size_t)n;
  float* h1   = ws + 2 * (size_t)n;
  float* h2   = h1 + (size_t)n * 64;

  const int tpb = 256;
  tag_init_deg<<<(n + tpb - 1) / tpb, tpb, 0, stream>>>(deg, n);
  tag_deg_scatter<<<(e + tpb - 1) / tpb, tpb, 0, stream>>>(dst, deg, e);
  tag_dinv<<<(n + tpb - 1) / tpb, tpb, 0, stream>>>(deg, dinv, n);

  const long long tsl = (long long)n * 16;
  const long long tsp = (long long)e * 16;
  // hop 1: x -> h1
  tag_selfloop<<<(int)((tsl + tpb - 1) / tpb), tpb, 0, stream>>>(x, dinv, h1, n);
  tag_spmm<<<(int)((tsp + tpb - 1) / tpb), tpb, 0, stream>>>(src, dst, dinv, x, h1, e);
  // hop 2: h1 -> h2
  tag_selfloop<<<(int)((tsl + tpb - 1) / tpb), tpb, 0, stream>>>(h1, dinv, h2, n);
  tag_spmm<<<(int)((tsp + tpb - 1) / tpb), tpb, 0, stream>>>(src, dst, dinv, h1, h2, e);

  // dense tail
  const int waves  = (n + 15) / 16;          // one 16-row tile per wave
  const int blocks = (waves + 7) / 8;        // 8 waves per 256-thread block
  tag_gemm<<<blocks, tpb, 0, stream>>>(x, h1, h2, W, b, out, n);
}